// GCN_28879360098390
// MI455X (gfx1250) — compile-verified
//
#include <hip/hip_runtime.h>
#include <math.h>

#define NN 50000
#define EE 800000
#define CC 128
#define HH 128
#define BB 256
#define CLS 10
#define MLPH 64
#define EPSV 1e-5f

typedef float v2f __attribute__((ext_vector_type(2)));
typedef float v8f __attribute__((ext_vector_type(8)));

// ---------------- degree / norm ----------------
__global__ void deg_init_k(float* deg, int n) {
    int i = blockIdx.x * blockDim.x + threadIdx.x;
    if (i < n) deg[i] = 1.0f;              // self-loop weight
}

__global__ void deg_accum_k(const int* __restrict__ dst, const float* __restrict__ ew,
                            float* deg, int e) {
    int i = blockIdx.x * blockDim.x + threadIdx.x;
    if (i < e) atomicAdd(&deg[dst[i]], ew[i]);
}

__global__ void dinv_k(float* deg, int n) {
    int i = blockIdx.x * blockDim.x + threadIdx.x;
    if (i < n) {
        float d = deg[i];
        deg[i] = d > 0.0f ? rsqrtf(d) : 0.0f;   // in-place: deg -> dinv
    }
}

// ---------------- generic zero ----------------
__global__ void zero_k(float* p, int n) {
    int i = blockIdx.x * blockDim.x + threadIdx.x;
    if (i < n) p[i] = 0.0f;
}

// ---------------- fp32 WMMA GEMM: Y[M x 128] = X[M x 128] @ W[128 x 128] ----------------
// One wave computes a 16-row strip (16x128 output) via 8 n-tile accumulators.
// W is staged transposed in LDS so B fragments are contiguous 8B ds loads.
__global__ void __launch_bounds__(128) gemm_wmma_k(const float* __restrict__ X,
                                                   const float* __restrict__ W,
                                                   float* __restrict__ Y, int mtiles) {
    __shared__ float wt[128 * 128];   // wt[n*128 + k] = W[k*128 + n]  (64 KB LDS)
    for (int i = threadIdx.x; i < 128 * 128; i += 128) {
        int k = i >> 7, n = i & 127;
        wt[n * 128 + k] = W[i];
    }
    __syncthreads();

    const int lane  = threadIdx.x & 31;
    const int wave  = threadIdx.x >> 5;
    const int mrow  = lane & 15;      // row-in-tile for A / col-in-tile for B,C
    const int khalf = lane >> 4;      // 0 or 1
    const int nwaves = gridDim.x * 4;

    for (int mt = blockIdx.x * 4 + wave; mt < mtiles; mt += nwaves) {
        const float* xrow = X + (size_t)(mt * 16 + mrow) * 128 + khalf * 2;
        v8f acc[8] = {};
        for (int kk = 0; kk < 128; kk += 4) {
            v2f a;
            a.x = xrow[kk];
            a.y = xrow[kk + 1];
#pragma unroll
            for (int nt = 0; nt < 8; ++nt) {
                const float* wp = wt + (nt * 16 + mrow) * 128 + kk + khalf * 2;
                v2f b;
                b.x = wp[0];
                b.y = wp[1];
                acc[nt] = __builtin_amdgcn_wmma_f32_16x16x4_f32(
                    false, a, false, b, (short)0, acc[nt], false, false);
            }
        }
#pragma unroll
        for (int nt = 0; nt < 8; ++nt) {
#pragma unroll
            for (int v = 0; v < 8; ++v) {
                int row = mt * 16 + v + khalf * 8;
                Y[(size_t)row * 128 + nt * 16 + mrow] = acc[nt][v];
            }
        }
    }
}

// ---------------- edge scatter: agg[dst] += norm * h[src] ----------------
// one wave per edge (grid-stride); lane handles 4 channels (float4 gather + 4 f32 atomics)
__global__ void scatter_k(const int* __restrict__ src, const int* __restrict__ dst,
                          const float* __restrict__ ew, const float* __restrict__ dinv,
                          const float* __restrict__ h, float* __restrict__ agg, int e) {
    int gid  = blockIdx.x * blockDim.x + threadIdx.x;
    int w    = gid >> 5;
    int lane = gid & 31;
    int nw   = (gridDim.x * blockDim.x) >> 5;
    for (int i = w; i < e; i += nw) {
        int s = src[i], d = dst[i];
        float nrm = dinv[s] * ew[i] * dinv[d];
        float4 v = ((const float4*)(h + (size_t)s * 128))[lane];
        float* ap = agg + (size_t)d * 128 + lane * 4;
        atomicAdd(ap + 0, nrm * v.x);
        atomicAdd(ap + 1, nrm * v.y);
        atomicAdd(ap + 2, nrm * v.z);
        atomicAdd(ap + 3, nrm * v.w);
    }
}

// ---------------- combine: agg = relu(agg + h*dinv^2 + bias) ----------------
__global__ void combine_k(float* __restrict__ agg, const float* __restrict__ h,
                          const float* __restrict__ dinv, const float* __restrict__ bias,
                          int n) {
    int i = blockIdx.x * blockDim.x + threadIdx.x;
    if (i >= n) return;
    int row = i >> 7, c = i & 127;
    float di = dinv[row];
    float v  = agg[i] + h[i] * di * di + bias[c];
    agg[i]   = v > 0.0f ? v : 0.0f;
}

// ---------------- BN stats: per-channel sum / sumsq ----------------
__global__ void __launch_bounds__(128) bnstats_k(const float* __restrict__ h,
                                                 float* sum, float* sumsq) {
    int c = threadIdx.x;               // 128 channels
    float s = 0.0f, s2 = 0.0f;
    for (int r = blockIdx.x; r < NN; r += gridDim.x) {
        float v = h[(size_t)r * 128 + c];
        s  += v;
        s2 += v * v;
    }
    atomicAdd(&sum[c], s);
    atomicAdd(&sumsq[c], s2);
}

__global__ void bnfinal_k(const float* sum, const float* sumsq,
                          const float* __restrict__ gamma, const float* __restrict__ beta,
                          float* scale, float* shift) {
    int c = threadIdx.x;
    if (c >= 128) return;
    float mu  = sum[c] * (1.0f / (float)NN);
    float var = sumsq[c] * (1.0f / (float)NN) - mu * mu;
    float inv = rsqrtf(var + EPSV);
    float sc  = gamma[c] * inv;
    scale[c]  = sc;
    shift[c]  = beta[c] - mu * sc;
}

// ---------------- fused BN-apply + global pool accumulate ----------------
__global__ void pool_k(const float* __restrict__ h, const int* __restrict__ batch,
                       const float* __restrict__ scale, const float* __restrict__ shift,
                       float* gsum, float* cnt) {
    int gid  = blockIdx.x * blockDim.x + threadIdx.x;
    int w    = gid >> 5;
    int lane = gid & 31;
    int nw   = (gridDim.x * blockDim.x) >> 5;
    float4 sc = ((const float4*)scale)[lane];
    float4 sh = ((const float4*)shift)[lane];
    for (int node = w; node < NN; node += nw) {
        int b = batch[node];
        float4 v = ((const float4*)(h + (size_t)node * 128))[lane];
        float* gp = gsum + (size_t)b * 128 + lane * 4;
        atomicAdd(gp + 0, v.x * sc.x + sh.x);
        atomicAdd(gp + 1, v.y * sc.y + sh.y);
        atomicAdd(gp + 2, v.z * sc.z + sh.z);
        atomicAdd(gp + 3, v.w * sc.w + sh.w);
        if (lane == 0) atomicAdd(&cnt[b], 1.0f);
    }
}

// ---------------- MLP head: one block per graph ----------------
__global__ void __launch_bounds__(64) head_k(const float* __restrict__ gsum,
                                             const float* __restrict__ cnt,
                                             const float* __restrict__ Wm1,
                                             const float* __restrict__ bm1,
                                             const float* __restrict__ Wm2,
                                             const float* __restrict__ bm2,
                                             float* __restrict__ out) {
    __shared__ float g[128];
    __shared__ float hid[MLPH];
    int b = blockIdx.x;
    int t = threadIdx.x;               // 64 threads
    float inv = 1.0f / fmaxf(cnt[b], 1.0f);
    for (int c = t; c < 128; c += 64) g[c] = gsum[(size_t)b * 128 + c] * inv;
    __syncthreads();
    float a = bm1[t];
    for (int c = 0; c < 128; ++c) a += g[c] * Wm1[c * MLPH + t];
    hid[t] = a;
    __syncthreads();
    if (t < CLS) {
        float o = bm2[t];
        for (int j = 0; j < MLPH; ++j) o += hid[j] * Wm2[j * CLS + t];
        out[b * CLS + t] = o;
    }
}

extern "C" void kernel_launch(void* const* d_in, const int* in_sizes, int n_in,
                              void* d_out, int out_size, void* d_ws, size_t ws_size,
                              hipStream_t stream) {
    const float* x     = (const float*)d_in[0];
    const int*   ei    = (const int*)d_in[1];
    const float* ew    = (const float*)d_in[2];
    const int*   batch = (const int*)d_in[3];
    const float* W1 = (const float*)d_in[4],  *b1 = (const float*)d_in[5];
    const float* W2 = (const float*)d_in[6],  *b2 = (const float*)d_in[7];
    const float* W3 = (const float*)d_in[8],  *b3 = (const float*)d_in[9];
    const float* gamma = (const float*)d_in[10], *beta = (const float*)d_in[11];
    const float* Wm1 = (const float*)d_in[12], *bm1 = (const float*)d_in[13];
    const float* Wm2 = (const float*)d_in[14], *bm2 = (const float*)d_in[15];
    float* out = (float*)d_out;

    const int* src = ei;
    const int* dst = ei + EE;

    // workspace layout (floats): bufH | bufA | dinv | stats(512) | gsum | cnt
    float* ws   = (float*)d_ws;
    float* bufH = ws;                               // N*H
    float* bufA = bufH + (size_t)NN * HH;           // N*H
    float* dinv = bufA + (size_t)NN * HH;           // N
    float* stats = dinv + NN;                       // sum|sumsq|scale|shift
    float* sum   = stats;
    float* sumsq = stats + 128;
    float* scale = stats + 256;
    float* shift = stats + 384;
    float* gsum  = stats + 512;                     // B*H
    float* cnt   = gsum + (size_t)BB * HH;          // B

    const int NH = NN * HH;
    const int mtiles = NN / 16;                     // 3125 exact

    // degree -> dinv
    deg_init_k<<<(NN + 255) / 256, 256, 0, stream>>>(dinv, NN);
    deg_accum_k<<<(EE + 255) / 256, 256, 0, stream>>>(dst, ew, dinv, EE);
    dinv_k<<<(NN + 255) / 256, 256, 0, stream>>>(dinv, NN);

    const float* layerW[3] = {W1, W2, W3};
    const float* layerB[3] = {b1, b2, b3};
    const float* in = x;
    for (int l = 0; l < 3; ++l) {
        gemm_wmma_k<<<(mtiles + 3) / 4, 128, 0, stream>>>(in, layerW[l], bufH, mtiles);
        zero_k<<<(NH + 255) / 256, 256, 0, stream>>>(bufA, NH);
        scatter_k<<<4096, 256, 0, stream>>>(src, dst, ew, dinv, bufH, bufA, EE);
        combine_k<<<(NH + 255) / 256, 256, 0, stream>>>(bufA, bufH, dinv, layerB[l], NH);
        in = bufA;
    }

    // BN + pool
    int statLen = 512 + BB * HH + BB;
    zero_k<<<(statLen + 255) / 256, 256, 0, stream>>>(stats, statLen);
    bnstats_k<<<512, 128, 0, stream>>>(bufA, sum, sumsq);
    bnfinal_k<<<1, 128, 0, stream>>>(sum, sumsq, gamma, beta, scale, shift);
    pool_k<<<2048, 256, 0, stream>>>(bufA, batch, scale, shift, gsum, cnt);

    // MLP head
    head_k<<<BB, 64, 0, stream>>>(gsum, cnt, Wm1, bm1, Wm2, bm2, out);
}